// Context_16080357556866
// MI455X (gfx1250) — compile-verified
//
#include <hip/hip_runtime.h>

#define DD 128
#define NSEG 32

typedef __attribute__((ext_vector_type(2))) float v2f;
typedef __attribute__((ext_vector_type(8))) float v8f;

// ---------------------------------------------------------------------------
// Kernel A: segment sums + counts. Per-wave private LDS accumulators so float4
// accumulation is race-free; one atomicAdd flush per block at the end.
// ---------------------------------------------------------------------------
__global__ void __launch_bounds__(128)
seg_sum_kernel(const float* __restrict__ x, const int* __restrict__ seg,
               int N, int rowsPerBlock,
               float* __restrict__ gS, float* __restrict__ gCnt) {
    __shared__ float lS[4 * NSEG * DD];   // 64 KB: [wave][seg][col]
    __shared__ float lC[4 * NSEG];
    const int tid  = threadIdx.x;
    const int lane = tid & 31;
    const int w    = tid >> 5;

    for (int i = tid; i < 4 * NSEG * DD; i += 128) lS[i] = 0.0f;
    for (int i = tid; i < 4 * NSEG; i += 128) lC[i] = 0.0f;
    __syncthreads();

    const int base = blockIdx.x * rowsPerBlock;
    int end = base + rowsPerBlock;
    if (end > N) end = N;

    for (int r = base + w; r < end; r += 4) {
        const int s = seg[r];                               // wave-uniform
        const float4 v = ((const float4*)(x + (size_t)r * DD))[lane];
        float* acc = &lS[(w * NSEG + s) * DD + lane * 4];
        acc[0] += v.x; acc[1] += v.y; acc[2] += v.z; acc[3] += v.w;
        if (lane == 0) lC[w * NSEG + s] += 1.0f;
    }
    __syncthreads();

    for (int i = tid; i < NSEG * DD; i += 128) {
        float t = lS[i] + lS[NSEG * DD + i] + lS[2 * NSEG * DD + i] + lS[3 * NSEG * DD + i];
        atomicAdd(&gS[i], t);
    }
    if (tid < NSEG) {
        float t = lC[tid] + lC[NSEG + tid] + lC[2 * NSEG + tid] + lC[3 * NSEG + tid];
        atomicAdd(&gCnt[tid], t);
    }
}

// ---------------------------------------------------------------------------
// One MLP layer via V_WMMA_F32_16X16X4_F32:  Y[32,128] = act(X[32,128] @ W + b)
// 16 output tiles of 16x16; 8 waves -> 2 tiles per wave; K-loop of 32 steps.
// A-frag (16x4 f32): lane<16 holds K=0(v0),K=1(v1); lane>=16 holds K=2,K=3.
// B-frag (4x16 f32): lane<16 holds rows K=0(v0),K=1(v1); lane>=16 rows K=2,K=3.
// C/D (16x16 f32):   VGPR v, lane<16 -> M=v; lane>=16 -> M=v+8.
// ACT: 0 = none, 1 = relu, 2 = sigmoid
// ---------------------------------------------------------------------------
template <int ACT>
__device__ __forceinline__ void mlp_layer(const float* Xs, const float* __restrict__ W,
                                          const float* __restrict__ b,
                                          float* Ys, float* __restrict__ gOut,
                                          int waveId, int lane) {
    const int l    = lane & 15;
    const int half = lane >> 4;
    for (int t = waveId * 2; t < waveId * 2 + 2; ++t) {
        const int mt = t >> 3;
        const int nt = t & 7;
        const int nCol = nt * 16 + l;
        const float bn = b[nCol];
        v8f acc;
#pragma unroll
        for (int v = 0; v < 8; ++v) acc[v] = bn;
#pragma unroll 4
        for (int kk = 0; kk < 32; ++kk) {
            const int k0 = kk * 4 + half * 2;
            v2f a, bb;
            a[0]  = Xs[(mt * 16 + l) * DD + k0];
            a[1]  = Xs[(mt * 16 + l) * DD + k0 + 1];
            bb[0] = W[k0 * DD + nCol];
            bb[1] = W[(k0 + 1) * DD + nCol];
            acc = __builtin_amdgcn_wmma_f32_16x16x4_f32(
                false, a, false, bb, (short)0, acc, false, false);
        }
#pragma unroll
        for (int v = 0; v < 8; ++v) {
            float y = acc[v];
            if (ACT == 1) y = fmaxf(y, 0.0f);
            if (ACT == 2) y = 1.0f / (1.0f + expf(-y));
            const int m = mt * 16 + half * 8 + v;
            Ys[m * DD + nCol] = y;
            if (gOut) gOut[m * DD + nCol] = y;
        }
    }
}

// ---------------------------------------------------------------------------
// Kernel B: single block. c_V = S/cnt; gate_V = MLP3_V(c_V); c_V2 = c_V*gate_V
// (scatter-mean shortcut: gate is segment-constant); gate_E = MLP3_E(c_V2).
// ---------------------------------------------------------------------------
__global__ void __launch_bounds__(256)
gate_mlp_kernel(const float* __restrict__ gS, const float* __restrict__ gCnt,
                const float* __restrict__ VW1, const float* __restrict__ Vb1,
                const float* __restrict__ VW2, const float* __restrict__ Vb2,
                const float* __restrict__ VW3, const float* __restrict__ Vb3,
                const float* __restrict__ EW1, const float* __restrict__ Eb1,
                const float* __restrict__ EW2, const float* __restrict__ Eb2,
                const float* __restrict__ EW3, const float* __restrict__ Eb3,
                float* __restrict__ gGateV, float* __restrict__ gGateE) {
    __shared__ float Xc[NSEG * DD];
    __shared__ float H0[NSEG * DD];
    __shared__ float H1[NSEG * DD];
    const int tid  = threadIdx.x;
    const int lane = tid & 31;
    const int w    = tid >> 5;

    for (int i = tid; i < NSEG * DD; i += 256) {
        const float c = gCnt[i >> 7];
        Xc[i] = gS[i] / fmaxf(c, 1.0f);
    }
    __syncthreads();

    mlp_layer<1>(Xc, VW1, Vb1, H0, nullptr, w, lane); __syncthreads();
    mlp_layer<1>(H0, VW2, Vb2, H1, nullptr, w, lane); __syncthreads();
    mlp_layer<2>(H1, VW3, Vb3, H0, gGateV,  w, lane); __syncthreads();

    // c_V2 = c_V * gate_V  (exactly equals scatter_mean of the gated nodes)
    for (int i = tid; i < NSEG * DD; i += 256) Xc[i] *= H0[i];
    __syncthreads();

    mlp_layer<1>(Xc, EW1, Eb1, H1, nullptr, w, lane); __syncthreads();
    mlp_layer<1>(H1, EW2, Eb2, H0, nullptr, w, lane); __syncthreads();
    mlp_layer<2>(H0, EW3, Eb3, H1, gGateE,  w, lane);
}

// ---------------------------------------------------------------------------
// Kernel C: one wave per node row; 32 lanes x float4 = one 128-float row.
// ---------------------------------------------------------------------------
__global__ void __launch_bounds__(256)
gate_nodes_kernel(const float* __restrict__ x, const int* __restrict__ seg,
                  const float* __restrict__ gate, float* __restrict__ out, int N) {
    const int lane = threadIdx.x & 31;
    const int row  = blockIdx.x * 8 + (threadIdx.x >> 5);
    if (row >= N) return;
    const int s = seg[row];                                  // wave-uniform
    const float4 v = ((const float4*)(x    + (size_t)row * DD))[lane];
    const float4 g = ((const float4*)(gate + (size_t)s   * DD))[lane];
    float4 o;
    o.x = v.x * g.x; o.y = v.y * g.y; o.z = v.z * g.z; o.w = v.w * g.w;
    ((float4*)(out + (size_t)row * DD))[lane] = o;
}

// ---------------------------------------------------------------------------
// Kernel D: one wave per edge row; seg = batch_id[edge_src[row]] (uniform).
// ---------------------------------------------------------------------------
__global__ void __launch_bounds__(256)
gate_edges_kernel(const float* __restrict__ ea, const int* __restrict__ edge_src,
                  const int* __restrict__ seg, const float* __restrict__ gate,
                  float* __restrict__ out, int E) {
    const int lane = threadIdx.x & 31;
    const int row  = blockIdx.x * 8 + (threadIdx.x >> 5);
    if (row >= E) return;
    const int s = seg[edge_src[row]];                        // wave-uniform
    const float4 v = ((const float4*)(ea   + (size_t)row * DD))[lane];
    const float4 g = ((const float4*)(gate + (size_t)s   * DD))[lane];
    float4 o;
    o.x = v.x * g.x; o.y = v.y * g.y; o.z = v.z * g.z; o.w = v.w * g.w;
    ((float4*)(out + (size_t)row * DD))[lane] = o;
}

extern "C" void kernel_launch(void* const* d_in, const int* in_sizes, int n_in,
                              void* d_out, int out_size, void* d_ws, size_t ws_size,
                              hipStream_t stream) {
    const float* dst_na   = (const float*)d_in[0];
    const float* ea       = (const float*)d_in[1];
    const int*   edge_idx = (const int*)d_in[2];   // [2,E]; row 0 = sources
    const int*   batch_id = (const int*)d_in[3];
    const float* VW1 = (const float*)d_in[4];
    const float* Vb1 = (const float*)d_in[5];
    const float* VW2 = (const float*)d_in[6];
    const float* Vb2 = (const float*)d_in[7];
    const float* VW3 = (const float*)d_in[8];
    const float* Vb3 = (const float*)d_in[9];
    const float* EW1 = (const float*)d_in[10];
    const float* Eb1 = (const float*)d_in[11];
    const float* EW2 = (const float*)d_in[12];
    const float* Eb2 = (const float*)d_in[13];
    const float* EW3 = (const float*)d_in[14];
    const float* Eb3 = (const float*)d_in[15];

    const int N = in_sizes[0] / DD;
    const int E = in_sizes[1] / DD;

    float* ws     = (float*)d_ws;
    float* gS     = ws;            // 4096 floats (32x128 segment sums)
    float* gCnt   = ws + 4096;     // 32 floats (padded to 128)
    float* gGateV = ws + 4224;     // 4096 floats
    float* gGateE = ws + 8320;     // 4096 floats

    // zero only the accumulators (harness poisons ws with 0xAA)
    hipMemsetAsync(d_ws, 0, (4096 + 128) * sizeof(float), stream);

    const int nBlocksA = 256;
    const int rowsPerBlock = (N + nBlocksA - 1) / nBlocksA;
    seg_sum_kernel<<<nBlocksA, 128, 0, stream>>>(dst_na, batch_id, N, rowsPerBlock, gS, gCnt);

    gate_mlp_kernel<<<1, 256, 0, stream>>>(gS, gCnt,
                                           VW1, Vb1, VW2, Vb2, VW3, Vb3,
                                           EW1, Eb1, EW2, Eb2, EW3, Eb3,
                                           gGateV, gGateE);

    float* outNodes = (float*)d_out;
    float* outEdges = outNodes + (size_t)N * DD;
    gate_nodes_kernel<<<(N + 7) / 8, 256, 0, stream>>>(dst_na, batch_id, gGateV, outNodes, N);
    gate_edges_kernel<<<(E + 7) / 8, 256, 0, stream>>>(ea, edge_idx, batch_id, gGateE, outEdges, E);
}